// ClassifierModel_28853590295098
// MI455X (gfx1250) — compile-verified
//
#include <hip/hip_runtime.h>
#include <hip/hip_bf16.h>
#include <stdint.h>

typedef __attribute__((ext_vector_type(2))) float v2f;
typedef __attribute__((ext_vector_type(8))) float v8f;

#define RS_SCALE   32.0f
#define RS_LOGEPS  1e-8f

__global__ __launch_bounds__(256)
void classifier_loss_kernel(const float* __restrict__ cls_scores,
                            const float* __restrict__ bbox_deltas,
                            const float* __restrict__ roi,
                            const float* __restrict__ labels,
                            const int*  __restrict__ neg_enabled_p,
                            float* __restrict__ out,
                            int P, int L)
{
    __shared__ __align__(16) float s_labels[128 * 4];   // 2048 B: labels[b]
    __shared__ float s_total;

    const int tid = threadIdx.x;
    const int b   = blockIdx.y;
    const int p   = blockIdx.x * 256 + tid;

    if (tid == 0) s_total = 0.0f;

    // ---- CDNA5 async copy: labels[b] (L*4 = 512 floats = 2KB) -> LDS ----
    // 256 lanes x 8 bytes each, one GLOBAL_LOAD_ASYNC_TO_LDS_B64 per lane.
    {
        const float* gsrc = labels + ((size_t)b * (size_t)L * 4) + (size_t)tid * 2;
        unsigned long long gaddr = (unsigned long long)(uintptr_t)gsrc;
        // HW truncates flat-shared address to its low 32 bits for LDS addressing.
        unsigned lds_addr = (unsigned)(uintptr_t)(&s_labels[tid * 2]);
        asm volatile("global_load_async_to_lds_b64 %0, %1, off"
                     :: "v"(lds_addr), "v"(gaddr)
                     : "memory");
        asm volatile("s_wait_asynccnt 0" ::: "memory");
    }

    // ---- per-proposal ROI (scaled to image coords); issue global loads early ----
    const float4 r4 = *reinterpret_cast<const float4*>(roi + ((size_t)(b * P + p)) * 4);
    const float s0 = cls_scores[(size_t)b * 2 * P + p];
    const float s1 = cls_scores[(size_t)b * 2 * P + P + p];
    const int   neg_i = *neg_enabled_p;

    const float rx = r4.x * RS_SCALE, ry = r4.y * RS_SCALE;
    const float rw = r4.z * RS_SCALE, rh = r4.w * RS_SCALE;
    const float rArea = rw * rh;
    const float rx2 = rx + rw, ry2 = ry + rh;

    __syncthreads();   // labels staged in LDS by all waves

    // ---- IoU argmax over L labels, DIVISION-FREE ----
    // iou_l > best  ⟺  inter_l * bestUni > bestInter * uni_l   (uni > 0)
    // Seed (bestInter=-1, bestUni=1) == "best iou = -1"; strict > keeps the
    // first-max index (jnp.argmax tie-breaking).
    float bestInter = -1.0f;
    float bestUni   =  1.0f;
    int   best_idx  =  0;
    const float4* lab4 = reinterpret_cast<const float4*>(s_labels);
#pragma unroll 4
    for (int l = 0; l < 128; ++l) {
        const float4 lb = lab4[l];                 // ds_load_b128 broadcast
        float ix = fminf(lb.x + lb.z, rx2) - fmaxf(lb.x, rx);
        float iy = fminf(lb.y + lb.w, ry2) - fmaxf(lb.y, ry);
        ix = fmaxf(ix, 0.0f);
        iy = fmaxf(iy, 0.0f);
        const float inter = ix * iy;
        const float uni   = lb.z * lb.w + rArea - inter;
        if (uni > 0.0f && inter * bestUni > bestInter * uni) {
            bestInter = inter;
            bestUni   = uni;
            best_idx  = l;
        }
    }
    // any_hit ⟺ max iou > 0 ⟺ bestInter > 0; pos = match > 0 (index 0 excluded!)
    const bool pos = (bestInter > 0.0f) && (best_idx > 0);

    // ---- classification term (2-way softmax CE) ----
    const float mx = fmaxf(s0, s1);
    const float e0 = __expf(s0 - mx), e1 = __expf(s1 - mx);
    const float inv = 1.0f / (e0 + e1);
    const float p0 = e0 * inv, p1 = e1 * inv;
    const float LOG_01 = -2.3025850929940457f;   // log(0.1)
    const float LOG_09 = -0.10536051565782628f;  // log(0.9)
    const float ce_pos = -(p0 * LOG_01 + p1 * LOG_09);
    const float ce_neg = -(p0 * LOG_09 + p1 * LOG_01);

    float per_prop;
    if (pos) {
        const float4 m4 = lab4[best_idx];          // matched label (unscaled)
        const float tx = (m4.x - rx) / rw;
        const float ty = (m4.y - ry) / rh;
        const float tw = __logf(fmaxf(m4.z / rw, RS_LOGEPS));
        const float th = __logf(fmaxf(m4.w / rh, RS_LOGEPS));
        const size_t bb = (size_t)b * 4 * P + p;
        const float d0 = bbox_deltas[bb + 0 * (size_t)P];
        const float d1 = bbox_deltas[bb + 1 * (size_t)P];
        const float d2 = bbox_deltas[bb + 2 * (size_t)P];
        const float d3 = bbox_deltas[bb + 3 * (size_t)P];
        float h = 0.0f, e, ae;
        e = tx - d0; ae = fabsf(e); h += (ae <= 1.0f) ? 0.5f * e * e : ae - 0.5f;
        e = ty - d1; ae = fabsf(e); h += (ae <= 1.0f) ? 0.5f * e * e : ae - 0.5f;
        e = tw - d2; ae = fabsf(e); h += (ae <= 1.0f) ? 0.5f * e * e : ae - 0.5f;
        e = th - d3; ae = fabsf(e); h += (ae <= 1.0f) ? 0.5f * e * e : ae - 0.5f;
        per_prop = 2.0f * (h * 0.25f) + ce_pos;
    } else {
        per_prop = (neg_i > 0) ? ce_neg : 0.0f;
    }

    // ---- wave32 sum reduction via V_WMMA_F32_16X16X4_F32 with ones-B ----
    // A[16x4]: lane's partial in K=0 (lanes 0-15) / K=2 (lanes 16-31); rest 0.
    // D = A x ones => every D[m][n] = rowsum(m). Lanes 0-15 sum rows 0-7,
    // lanes 16-31 sum rows 8-15, so lane0 + lane16 = full 32-lane total.
    float sum8;
#if __has_builtin(__builtin_amdgcn_wmma_f32_16x16x4_f32)
    {
        v2f a;  a.x = per_prop; a.y = 0.0f;
        v2f b1; b1.x = 1.0f;    b1.y = 1.0f;
        v8f c = {};
        c = __builtin_amdgcn_wmma_f32_16x16x4_f32(
                /*neg_a=*/false, a, /*neg_b=*/false, b1,
                /*c_mod=*/(short)0, c, /*reuse_a=*/false, /*reuse_b=*/false);
        sum8 = c[0] + c[1] + c[2] + c[3] + c[4] + c[5] + c[6] + c[7];
    }
    const unsigned lane = tid & 31u;
    if ((lane & 15u) == 0u) atomicAdd(&s_total, sum8);   // lanes 0 and 16
#else
    sum8 = per_prop;
    for (int off = 16; off > 0; off >>= 1)
        sum8 += __shfl_down(sum8, off, 32);
    if ((tid & 31u) == 0u) atomicAdd(&s_total, sum8);
#endif

    __syncthreads();
    if (tid == 0) atomicAdd(out, s_total);
}

extern "C" void kernel_launch(void* const* d_in, const int* in_sizes, int n_in,
                              void* d_out, int out_size, void* d_ws, size_t ws_size,
                              hipStream_t stream) {
    const float* cls    = (const float*)d_in[0];   // [B, 2P]
    const float* bbox   = (const float*)d_in[1];   // [B, 4P]
    const float* roi    = (const float*)d_in[2];   // [B, P, 4]
    const float* labels = (const float*)d_in[3];   // [B, L, 4]
    const int*   neg    = (const int*)d_in[4];     // scalar
    float* out = (float*)d_out;

    const int B = 128;
    const int P = in_sizes[0] / (B * 2);           // 4096
    const int L = in_sizes[3] / (B * 4);           // 128

    hipMemsetAsync(out, 0, sizeof(float), stream); // graph-capture safe
    dim3 grid(P / 256, B);
    classifier_loss_kernel<<<grid, 256, 0, stream>>>(cls, bbox, roi, labels, neg, out, P, L);
}